// EvolutionAgentQuantum_83442624627033
// MI455X (gfx1250) — compile-verified
//
#include <hip/hip_runtime.h>
#include <hip/hip_bf16.h>

// ---------------------------------------------------------------------------
// 10-qubit statevector sim, one wave32 per sample.
// State: 1024 complex amps -> 32 float2 per lane (64 VGPRs), register-resident.
// amp index k = lane*32 + j ; bit b of k: b<5 -> local bit of j, b>=5 -> lane bit.
// qubit q <-> bit (9-q).
// Cross-lane xor exchanges use DPP8 / DPP16(row_ror:8) / v_permlanex16 —
// pure VALU, no LDS traffic, no DScnt waits (vs ds_swizzle from __shfl_xor).
// Fusion: input RZ_i and weight RZ_i commute past intervening gates -> merged.
// Readout: <Z_0..3> depend on lane bits only -> per-lane prob pre-reduction,
// then O(8x4) = P(8x32) x S(32x4) per block via v_wmma_f32_16x16x4_f32.
// ---------------------------------------------------------------------------

typedef __attribute__((ext_vector_type(2))) float v2f;
typedef __attribute__((ext_vector_type(8))) float v8f;

#define NQ 10
#define NL 5
#define WAVES_PER_BLOCK 8

// ---- VALU lane-xor exchange (all lanes active at every call site) ----------
constexpr unsigned dpp8_sel_xor(int m) {
    unsigned s = 0;
    for (int i = 0; i < 8; ++i) s |= (unsigned)((i ^ m) & 7) << (3 * i);
    return s;
}

template <int LM>
__device__ __forceinline__ float lane_xor(float v) {
    if constexpr (LM < 8) {
        // xor within 8-lane group: v_mov_b32 dpp8
        constexpr unsigned sel = dpp8_sel_xor(LM);
        return __int_as_float(
            __builtin_amdgcn_mov_dpp8(__float_as_int(v), (int)sel));
    } else if constexpr (LM == 8) {
        // xor8 within a 16-lane row == row_ror:8 (DPP16 ctrl 0x128)
        return __int_as_float(__builtin_amdgcn_update_dpp(
            0, __float_as_int(v), 0x128, 0xF, 0xF, true));
    } else {
        // xor16: cross-half identity permute via v_permlanex16_b32
        int iv = __float_as_int(v);
        return __int_as_float(__builtin_amdgcn_permlanex16(
            iv, iv, 0x76543210, (int)0xFEDCBA98, false, false));
    }
}

// ---- RY on qubit with index-bit B; c=cos(theta/2), s=sin(theta/2) ----------
template <int B>
__device__ __forceinline__ void apply_ry(float2 (&st)[32], unsigned lane, float c, float s) {
    if constexpr (B < 5) {
        constexpr int m = 1 << B;
#pragma unroll
        for (int j = 0; j < 32; ++j) {
            if ((j & m) == 0) {
                const int j1 = j | m;
                float2 a0 = st[j], a1 = st[j1];
                st[j].x  = c * a0.x - s * a1.x;
                st[j].y  = c * a0.y - s * a1.y;
                st[j1].x = s * a0.x + c * a1.x;
                st[j1].y = s * a0.y + c * a1.y;
            }
        }
    } else {
        constexpr int lm = 1 << (B - 5);
        // bit==0: new = c*mine - s*other ; bit==1: new = c*mine + s*other
        const float sgn = (lane & lm) ? s : -s;
#pragma unroll
        for (int j = 0; j < 32; ++j) {
            float ox = lane_xor<lm>(st[j].x);
            float oy = lane_xor<lm>(st[j].y);
            st[j].x = c * st[j].x + sgn * ox;
            st[j].y = c * st[j].y + sgn * oy;
        }
    }
}

// ---- RZ on qubit with index-bit B: bit?e^{+it}:e^{-it}, c=cos t, s=sin t ---
template <int B>
__device__ __forceinline__ void apply_rz(float2 (&st)[32], unsigned lane, float c, float s) {
    if constexpr (B < 5) {
        constexpr int m = 1 << B;
#pragma unroll
        for (int j = 0; j < 32; ++j) {
            const float sp = (j & m) ? s : -s;
            float x = st[j].x, y = st[j].y;
            st[j].x = x * c - y * sp;
            st[j].y = x * sp + y * c;
        }
    } else {
        constexpr int lm = 1 << (B - 5);
        const float sp = (lane & lm) ? s : -s;
#pragma unroll
        for (int j = 0; j < 32; ++j) {
            float x = st[j].x, y = st[j].y;
            st[j].x = x * c - y * sp;
            st[j].y = x * sp + y * c;
        }
    }
}

// ---- CNOT: control index-bit BC, target index-bit BT -----------------------
template <int BC, int BT>
__device__ __forceinline__ void apply_cnot(float2 (&st)[32], unsigned lane) {
    if constexpr (BT < 5) {
        constexpr int tm = 1 << BT;
        if constexpr (BC < 5) {
            constexpr int cm = 1 << BC;
#pragma unroll
            for (int j = 0; j < 32; ++j) {
                if ((j & cm) && !(j & tm)) {
                    float2 t = st[j]; st[j] = st[j | tm]; st[j | tm] = t;
                }
            }
        } else {
            constexpr int clm = 1 << (BC - 5);
            const bool ctrl = (lane & clm) != 0;
#pragma unroll
            for (int j = 0; j < 32; ++j) {
                if (!(j & tm)) {
                    float2 a = st[j], b = st[j | tm];
                    st[j]      = ctrl ? b : a;
                    st[j | tm] = ctrl ? a : b;
                }
            }
        }
    } else {
        constexpr int tlm = 1 << (BT - 5);
        if constexpr (BC < 5) {
            constexpr int cm = 1 << BC;
#pragma unroll
            for (int j = 0; j < 32; ++j) {
                if (j & cm) {   // compile-time condition; whole wave exchanges
                    st[j].x = lane_xor<tlm>(st[j].x);
                    st[j].y = lane_xor<tlm>(st[j].y);
                }
            }
        } else {
            constexpr int clm = 1 << (BC - 5);
            const bool ctrl = (lane & clm) != 0;
#pragma unroll
            for (int j = 0; j < 32; ++j) {
                float ox = lane_xor<tlm>(st[j].x);
                float oy = lane_xor<tlm>(st[j].y);
                if (ctrl) { st[j].x = ox; st[j].y = oy; }
            }
        }
    }
}

// ---- per-layer gate drivers (recursive templates keep bit positions const) -
// RY_i(isc[i]*x_i) ; fused RZ_i(isc[i+10]*x_i + w[i])
template <int I>
__device__ __forceinline__ void input_rots(float2 (&st)[32], unsigned lane,
                                           const float* __restrict__ iscRow,
                                           const float* __restrict__ wRow,
                                           const float* __restrict__ xr) {
    if constexpr (I < NQ) {
        const float xv = xr[I];
        float s, c, t;
        t = 0.5f * (iscRow[I] * xv);
        __sincosf(t, &s, &c);
        apply_ry<9 - I>(st, lane, c, s);
        t = 0.5f * (iscRow[I + NQ] * xv + wRow[I]);   // fused RZ
        __sincosf(t, &s, &c);
        apply_rz<9 - I>(st, lane, c, s);
        input_rots<I + 1>(st, lane, iscRow, wRow, xr);
    }
}

template <int I>
__device__ __forceinline__ void weight_ry(float2 (&st)[32], unsigned lane,
                                          const float* __restrict__ wRow) {
    if constexpr (I < NQ) {
        float s, c;
        __sincosf(0.5f * wRow[I + NQ], &s, &c);
        apply_ry<9 - I>(st, lane, c, s);
        weight_ry<I + 1>(st, lane, wRow);
    }
}

__device__ __forceinline__ void cnot_ring(float2 (&st)[32], unsigned lane) {
    // CNOT(q, q+1 mod 10): bits (9-q, 9-(q+1)%10)
    apply_cnot<9, 8>(st, lane);
    apply_cnot<8, 7>(st, lane);
    apply_cnot<7, 6>(st, lane);
    apply_cnot<6, 5>(st, lane);
    apply_cnot<5, 4>(st, lane);
    apply_cnot<4, 3>(st, lane);
    apply_cnot<3, 2>(st, lane);
    apply_cnot<2, 1>(st, lane);
    apply_cnot<1, 0>(st, lane);
    apply_cnot<0, 9>(st, lane);
}

// ---- sign matrix element S[k][n] = (+-1) for n<4, else 0 -------------------
__device__ __forceinline__ float sign_elem(int k, int n) {
    return (n < 4) ? (((k >> (4 - n)) & 1) ? -1.0f : 1.0f) : 0.0f;
}

__global__ __launch_bounds__(32 * WAVES_PER_BLOCK)
void EvolutionAgentQuantum_kernel(const float* __restrict__ x,
                                  const float* __restrict__ input_scaling,
                                  const float* __restrict__ weights,
                                  const float* __restrict__ output_scaling,
                                  float* __restrict__ out) {
    const unsigned lane = threadIdx.x & 31u;
    const unsigned wave = threadIdx.x >> 5;
    const unsigned sample = blockIdx.x * WAVES_PER_BLOCK + wave;
    const float* xr = x + sample * NQ;

    // |0...0>
    float2 st[32];
#pragma unroll
    for (int j = 0; j < 32; ++j) { st[j].x = 0.0f; st[j].y = 0.0f; }
    st[0].x = (lane == 0) ? 1.0f : 0.0f;

    for (int layer = 0; layer < NL; ++layer) {
        const float* iscRow = input_scaling + layer * (2 * NQ);
        const float* wRow   = weights       + layer * (2 * NQ);
        input_rots<0>(st, lane, iscRow, wRow, xr);  // RY + fused RZ per qubit
        weight_ry<0>(st, lane, wRow);
        cnot_ring(st, lane);
    }

    // per-lane probability pre-reduction (measured qubits 0..3 are lane bits 4..1)
    float p = 0.0f;
#pragma unroll
    for (int j = 0; j < 32; ++j) p += st[j].x * st[j].x + st[j].y * st[j].y;

    // 16-row tile, rows 8..15 zero-padded: WMMA A-loads stay unpredicated.
    __shared__ __align__(16) float ptile[16][32];
    ptile[wave + 8][lane] = 0.0f;
    ptile[wave][lane]     = p;
    __syncthreads();

    // ---- readout GEMM: O(8x4) = P(8x32) x S(32x4) via v_wmma_f32_16x16x4 ---
    if (wave == 0) {  // wave-uniform branch: EXEC stays all-ones inside
        const int m    = (int)(lane & 15u);
        const int half = (int)(lane >> 4);   // lanes 16-31 carry K+2 per A/B layout

        // Hoist all A chunks (ds_load_b64, no exec masking) ...
        const float2* prow = reinterpret_cast<const float2*>(&ptile[m][0]);
        v2f a[8];
#pragma unroll
        for (int kc = 0; kc < 8; ++kc) {
            float2 av = prow[kc * 2 + half];   // = {P[m][kc*4+half*2], P[m][..+1]}
            a[kc].x = av.x;
            a[kc].y = av.y;
        }
        // ... and all B (sign-matrix) chunks before the WMMA chain.
        v2f b[8];
#pragma unroll
        for (int kc = 0; kc < 8; ++kc) {
            const int k0 = kc * 4 + half * 2;
            b[kc].x = sign_elem(k0,     m);
            b[kc].y = sign_elem(k0 + 1, m);
        }

        v8f acc = {};
#pragma unroll
        for (int kc = 0; kc < 8; ++kc)
            acc = __builtin_amdgcn_wmma_f32_16x16x4_f32(
                false, a[kc], false, b[kc], (short)0, acc, false, false);

        // D layout: VGPR r, lanes 0-15 -> D[r][lane]; samples are rows 0..7
        if (lane < 4) {
            const float os = output_scaling[lane];
            float* o = out + (size_t)blockIdx.x * WAVES_PER_BLOCK * 4 + lane;
#pragma unroll
            for (int r = 0; r < WAVES_PER_BLOCK; ++r)
                o[r * 4] = acc[r] * os;
        }
    }
}

extern "C" void kernel_launch(void* const* d_in, const int* in_sizes, int n_in,
                              void* d_out, int out_size, void* d_ws, size_t ws_size,
                              hipStream_t stream) {
    const float* x   = (const float*)d_in[0];   // (B, 10)
    const float* isc = (const float*)d_in[1];   // (5, 20)
    const float* w   = (const float*)d_in[2];   // (5, 20)
    const float* os  = (const float*)d_in[3];   // (4,)
    float* out = (float*)d_out;                 // (B, 4)

    const int batch  = in_sizes[0] / NQ;                  // 4096
    const int blocks = batch / WAVES_PER_BLOCK;           // 512
    EvolutionAgentQuantum_kernel<<<blocks, 32 * WAVES_PER_BLOCK, 0, stream>>>(
        x, isc, w, os, out);
}